// CovarianceLayer_4337916969511
// MI455X (gfx1250) — compile-verified
//
#include <hip/hip_runtime.h>

// CovarianceLayer for MI455X (gfx1250): memory-bound fp32 separable stencil.
// ~200 MB HBM traffic vs <1 GFLOP -> bandwidth-bound. CDNA5 paths used:
//  - Tensor Data Mover stages 40x40 halo tiles (global->LDS, TENSORcnt)
//  - final vertical 5-tap box sum = banded matmul via V_WMMA_F32_16X16X4_F32

#define IMG   1024
#define OUTD  1016        // IMG - 8
#define TILE  32          // output tile edge
#define INT   40          // input tile edge (TILE + 8)
#define MIDT  36          // mid-grid tile edge (TILE + 4)
#define STR   40          // LDS row stride (TDM writes rows contiguously)

typedef unsigned int v4u __attribute__((ext_vector_type(4)));
typedef int          v4i __attribute__((ext_vector_type(4)));
typedef int          v8i __attribute__((ext_vector_type(8)));
typedef float        v2f __attribute__((ext_vector_type(2)));
typedef float        v8f __attribute__((ext_vector_type(8)));

#define HAVE_TDM  __has_builtin(__builtin_amdgcn_tensor_load_to_lds)
#define HAVE_WMMA __has_builtin(__builtin_amdgcn_wmma_f32_16x16x4_f32)

__device__ __forceinline__ unsigned lds_off(const void* p) {
  // Flat shared address = {aperture_hi, lds_byte_offset}; low 32 bits are the
  // wave-relative LDS byte address.
  return (unsigned)(size_t)p;
}

#if HAVE_TDM
// One TDM descriptor moves a 40x40 fp32 tile (row stride = IMG floats) into
// LDS. tensor_dim0/1 are the remaining extents from the tile origin so halo
// reads past the image edge return zero (hardware OOB) instead of faulting.
__device__ __forceinline__ void tdm_load_tile(unsigned lds_byte, const float* gtile,
                                              unsigned rem_cols, unsigned rem_rows) {
  unsigned long long ga = (unsigned long long)(size_t)gtile;
  v4u g0;
  g0[0] = 1u;                                     // count=1, user descriptor
  g0[1] = lds_byte;                               // lds_addr
  g0[2] = (unsigned)ga;                           // global_addr[31:0]
  g0[3] = (unsigned)(ga >> 32) | 0x80000000u;     // global_addr[56:32] | type=2
  v8i g1;
  g1[0] = 0x00020000;                             // data_size=2 (4 bytes); no mask/pad
  g1[1] = (int)((rem_cols & 0xFFFFu) << 16);      // tensor_dim0[15:0]
  g1[2] = (int)(((rem_cols >> 16) & 0xFFFFu) | ((rem_rows & 0xFFFFu) << 16));
  g1[3] = (int)(((rem_rows >> 16) & 0xFFFFu) | (((unsigned)INT) << 16)); // tile_dim0=40
  g1[4] = INT;                                    // tile_dim1=40, tile_dim2=0
  g1[5] = IMG;                                    // tensor_dim0_stride[31:0]
  g1[6] = 0;                                      // stride0 hi | stride1 lo
  g1[7] = (IMG * IMG) >> 16;                      // tensor_dim1_stride hi (unused for 2D)
  v4i gz; gz[0] = 0; gz[1] = 0; gz[2] = 0; gz[3] = 0;   // 2D tensor: groups 2/3 zero
#if __clang_major__ >= 23
  v8i gz8; for (int i = 0; i < 8; ++i) gz8[i] = 0;
  __builtin_amdgcn_tensor_load_to_lds(g0, g1, gz, gz, gz8, 0);
#else
  __builtin_amdgcn_tensor_load_to_lds(g0, g1, gz, gz, 0);
#endif
}
#endif

__global__ __launch_bounds__(256)
void cov5x5_kernel(const float* __restrict__ x,
                   const float* __restrict__ y,
                   float* __restrict__ out) {
  __shared__ __align__(16) float Xs[INT * STR];   // raw x tile (40x40)
  __shared__ __align__(16) float Ys[INT * STR];   // raw y tile
  __shared__ float Hx[INT * STR];                 // horizontal 5-sums of x (40x36 valid)
  __shared__ float Hy[INT * STR];
  __shared__ float P [MIDT * STR];                // centered product (36x36 valid)
  __shared__ float HP[MIDT * STR];                // horizontal 5-sums of P (36x32 valid)

  const int tid = threadIdx.x;
  const int gx0 = blockIdx.x * TILE;
  const int gy0 = blockIdx.y * TILE;
  const int b   = blockIdx.z;

  const float* xb = x + (size_t)b * IMG * IMG;
  const float* yb = y + (size_t)b * IMG * IMG;

  // ---- Phase A: stage 40x40 halo tiles of x and y into LDS ----------------
#if HAVE_TDM
  if (tid < 32) {                                 // wave 0 issues both TDM ops
    const unsigned rem_cols = (unsigned)(IMG - gx0);
    const unsigned rem_rows = (unsigned)(IMG - gy0);
    tdm_load_tile(lds_off(Xs), xb + (size_t)gy0 * IMG + gx0, rem_cols, rem_rows);
    tdm_load_tile(lds_off(Ys), yb + (size_t)gy0 * IMG + gx0, rem_cols, rem_rows);
    __builtin_amdgcn_s_wait_tensorcnt(0);
  }
  __syncthreads();
#else
  // Fallback: per-lane async global->LDS B128 transfers (ASYNCcnt path).
  for (int l = tid; l < 2 * INT * 10; l += 256) {
    const int isY   = (l >= INT * 10);
    const int rem   = isY ? (l - INT * 10) : l;
    const int r     = rem / 10;
    const int chunk = rem % 10;
    int grow = gy0 + r;         if (grow > IMG - 1) grow = IMG - 1;   // clamp: only
    int gcol = gx0 + chunk * 4; if (gcol > IMG - 4) gcol = IMG - 4;   // feeds masked outputs
    const float* gp = (isY ? yb : xb) + (size_t)grow * IMG + gcol;
    float* lp = (isY ? Ys : Xs) + r * STR + chunk * 4;
    asm volatile("global_load_async_to_lds_b128 %0, %1, off"
                 :: "v"(lds_off(lp)), "v"(gp) : "memory");
  }
  asm volatile("s_wait_asynccnt 0" ::: "memory");
  __syncthreads();
#endif

  // ---- Phase B: horizontal 5-sums of raw tiles (40 rows x 36 cols, x & y) -
  for (int e = tid; e < 2 * INT * MIDT; e += 256) {
    const int isY = (e >= INT * MIDT);
    const int rem = isY ? (e - INT * MIDT) : e;
    const int r = rem / MIDT, c = rem % MIDT;
    const float* S = (isY ? Ys : Xs) + r * STR + c;
    const float s = S[0] + S[1] + S[2] + S[3] + S[4];
    (isY ? Hy : Hx)[r * STR + c] = s;
  }
  __syncthreads();

  // ---- Phase C: centered product on 36x36 mid-grid ------------------------
  for (int e = tid; e < MIDT * MIDT; e += 256) {
    const int r = e / MIDT, c = e % MIDT;
    const float* hx = Hx + r * STR + c;
    const float* hy = Hy + r * STR + c;
    const float mx = (hx[0] + hx[STR] + hx[2 * STR] + hx[3 * STR] + hx[4 * STR]) * 0.04f;
    const float my = (hy[0] + hy[STR] + hy[2 * STR] + hy[3 * STR] + hy[4 * STR]) * 0.04f;
    const float xc = Xs[(r + 2) * STR + (c + 2)] - mx;
    const float yc = Ys[(r + 2) * STR + (c + 2)] - my;
    P[r * STR + c] = xc * yc;
  }
  __syncthreads();

  // ---- Phase D: horizontal 5-sums of P (36 rows x 32 cols) ----------------
  for (int e = tid; e < MIDT * TILE; e += 256) {
    const int r = e / TILE, j = e % TILE;
    const float* p = P + r * STR + j;
    HP[r * STR + j] = p[0] + p[1] + p[2] + p[3] + p[4];
  }
  __syncthreads();

  // ---- Phase E: vertical 5-sum * 1/25 -> 32x32 outputs --------------------
#if HAVE_WMMA
  // out(32x32) = Band(32x36) x HP(36x32), Band[i][k] = 0.04*[i <= k <= i+4].
  // Four 16x16 blocks, one per wave (waves 0..3, full EXEC), K=36 in 9 slices
  // of V_WMMA_F32_16X16X4_F32 with accumulator chaining (fp32-exact path).
  {
    const int wave = tid >> 5;
    const int lane = tid & 31;
    if (wave < 4) {
      const int R  = (wave >> 1) << 4;            // output row block: 0 or 16
      const int N0 = (wave & 1) << 4;             // output col block: 0 or 16
      const int h  = lane >> 4;                   // lane half (K phase)
      const int m  = lane & 15;                   // A row (lanes) == D column
      v8f acc = {};
      for (int kk = 0; kk < MIDT; kk += 4) {
        const int k0 = kk + h * 2;                // K held by this half in VGPR0
        v2f a, bv;
        // A (16x4): VGPR0 = K=k0, VGPR1 = K=k0+1; band weight 1/25 inside band
        a.x = (k0     >= R + m && k0     <= R + m + 4) ? 0.04f : 0.0f;
        a.y = (k0 + 1 >= R + m && k0 + 1 <= R + m + 4) ? 0.04f : 0.0f;
        // B (4x16): VGPR0 = row k0, VGPR1 = row k0+1, column = m
        bv.x = HP[k0       * STR + N0 + m];
        bv.y = HP[(k0 + 1) * STR + N0 + m];
        acc = __builtin_amdgcn_wmma_f32_16x16x4_f32(
            /*neg_a=*/false, a, /*neg_b=*/false, bv,
            /*c_mod=*/(short)0, acc, /*reuse_a=*/false, /*reuse_b=*/false);
      }
      // D (16x16): VGPR r = rows {R+r, R+r+8} (by lane half), column N0+m.
      float* ob = out + (size_t)b * OUTD * OUTD;
      const int gj = gx0 + N0 + m;
#pragma unroll
      for (int r = 0; r < 8; ++r) {
        const int gi = gy0 + R + r + h * 8;
        if (gi < OUTD && gj < OUTD) ob[(size_t)gi * OUTD + gj] = acc[r];
      }
    }
  }
#else
  for (int e = tid; e < TILE * TILE; e += 256) {
    const int i = e / TILE, j = e % TILE;
    const int gi = gy0 + i, gj = gx0 + j;
    if (gi < OUTD && gj < OUTD) {
      const float* hp = HP + i * STR + j;
      const float v = (hp[0] + hp[STR] + hp[2 * STR] + hp[3 * STR] + hp[4 * STR]) * 0.04f;
      out[(size_t)b * OUTD * OUTD + (size_t)gi * OUTD + gj] = v;
    }
  }
#endif
}

extern "C" void kernel_launch(void* const* d_in, const int* in_sizes, int n_in,
                              void* d_out, int out_size, void* d_ws, size_t ws_size,
                              hipStream_t stream) {
  const float* x = (const float*)d_in[0];
  const float* y = (const float*)d_in[1];
  // d_in[2] (mean_mask) and d_in[3] (ones_mask) are compile-time-known kernels
  // (1/25 box and center delta) — baked into the stencil.
  float* out = (float*)d_out;

  const int B = in_sizes[0] / (IMG * IMG);           // 16
  const int tiles = (OUTD + TILE - 1) / TILE;        // 32
  dim3 grid(tiles, tiles, B);
  dim3 block(256);
  hipLaunchKernelGGL(cov5x5_kernel, grid, block, 0, stream, x, y, out);
}